// MultiDDINOLoss_12841952215053
// MI455X (gfx1250) — compile-verified
//
#include <hip/hip_runtime.h>
#include <hip/hip_bf16.h>
#include <math.h>

typedef __attribute__((ext_vector_type(2))) float v2f;
typedef __attribute__((ext_vector_type(8))) float v8f;

#define D_DIM   4096
#define F_DIM   256
#define B_SZ    32
#define P_GL    196
#define P_LC    36
#define T_ROWS  12544   // 2*B*P
#define S_ROWS  21760
#define HALF_T  6272    // B*P
#define LOC_ROWS 1152   // B*PL
#define N_PAIRS 30976   // 2*HALF_T + 16*LOC_ROWS
#define NEG_INF (-3.0e38f)

__device__ __forceinline__ float get_ttemp(int e) {
    // np.linspace(0.04, 0.07, 30) then 0.07
    return (e >= 30) ? 0.07f : (0.04f + (0.03f / 29.0f) * (float)e);
}

__device__ __forceinline__ float blockReduceSum(float v, float* sdata) {
    int tid = threadIdx.x;
    sdata[tid] = v; __syncthreads();
    for (int s = 128; s > 0; s >>= 1) {
        if (tid < s) sdata[tid] += sdata[tid + s];
        __syncthreads();
    }
    float r = sdata[0]; __syncthreads();
    return r;
}

__device__ __forceinline__ float blockReduceMax(float v, float* sdata) {
    int tid = threadIdx.x;
    sdata[tid] = v; __syncthreads();
    for (int s = 128; s > 0; s >>= 1) {
        if (tid < s) sdata[tid] = fmaxf(sdata[tid], sdata[tid + s]);
        __syncthreads();
    }
    float r = sdata[0]; __syncthreads();
    return r;
}

// term id -> (view, teacher crop, view size, student row offset, pair offset)
__device__ __forceinline__ void decode_term(int t, int& v, int& iq, int& sz,
                                            int& soff, int& poff) {
    if (t == 0)      { v = 0; iq = 1; }
    else if (t == 1) { v = 1; iq = 0; }
    else             { v = 2 + (t - 2) / 2; iq = (t - 2) & 1; }
    sz   = (v < 2) ? P_GL : P_LC;
    soff = (v < 2) ? v * HALF_T : 2 * HALF_T + (v - 2) * LOC_ROWS;
    poff = (t < 2) ? t * HALF_T : 2 * HALF_T + (t - 2) * LOC_ROWS;
}

// ---- teacher feature inverse norms: invn = 1/max(||x||, 1e-12) ----
__global__ void tfea_invnorm_kernel(const float* __restrict__ t_fea,
                                    float* __restrict__ invn) {
    __shared__ float sdata[256];
    int row = blockIdx.x;
    float x = t_fea[(size_t)row * F_DIM + threadIdx.x];
    float s = blockReduceSum(x * x, sdata);
    if (threadIdx.x == 0) invn[row] = 1.0f / fmaxf(sqrtf(s), 1e-12f);
}

// ---- teacher row softmax stats: m = max((t-c)/T), invZ = 1/sum(exp(..-m)) ----
__global__ void tstats_kernel(const float* __restrict__ t, const float* __restrict__ center,
                              const int* __restrict__ epoch,
                              float* __restrict__ m_out, float* __restrict__ invZ_out) {
    __shared__ float sdata[256];
    int row = blockIdx.x, tid = threadIdx.x;
    float invT = 1.0f / get_ttemp(*epoch);
    const float* trow = t + (size_t)row * D_DIM;
    float v[16];
    float lmax = NEG_INF;
#pragma unroll
    for (int i = 0; i < 16; ++i) {
        int d = tid + i * 256;
        v[i] = (trow[d] - center[d]) * invT;
        lmax = fmaxf(lmax, v[i]);
    }
    float m = blockReduceMax(lmax, sdata);
    float ls = 0.f;
#pragma unroll
    for (int i = 0; i < 16; ++i) ls += expf(v[i] - m);
    float Z = blockReduceSum(ls, sdata);
    if (tid == 0) { m_out[row] = m; invZ_out[row] = 1.0f / Z; }
}

// ---- student row logsumexp of 10*s (streamed non-temporally: 356MB > L2) ----
__global__ void slse_kernel(const float* __restrict__ s, float* __restrict__ lse) {
    __shared__ float sdata[256];
    int row = blockIdx.x, tid = threadIdx.x;
    const float* sr = s + (size_t)row * D_DIM;
    float v[16];
    float lmax = NEG_INF;
#pragma unroll
    for (int i = 0; i < 16; ++i) {
        v[i] = __builtin_nontemporal_load(sr + tid + i * 256) * 10.0f;
        lmax = fmaxf(lmax, v[i]);
    }
    float m = blockReduceMax(lmax, sdata);
    float zz = 0.f;
#pragma unroll
    for (int i = 0; i < 16; ++i) zz += expf(v[i] - m);
    float Z = blockReduceSum(zz, sdata);
    if (tid == 0) lse[row] = m + logf(Z);
}

// ---- deterministic column-sum of t_region_out (8 row chunks) ----
__global__ void center_partial_kernel(const float* __restrict__ t, float* __restrict__ part) {
    int d = blockIdx.x * 256 + threadIdx.x;
    int chunk = blockIdx.y;
    int r0 = chunk * (T_ROWS / 8), r1 = r0 + (T_ROWS / 8);
    float s = 0.f;
    for (int r = r0; r < r1; ++r) s += t[(size_t)r * D_DIM + d];
    part[chunk * D_DIM + d] = s;
}

__global__ void center_final_kernel(const float* __restrict__ part,
                                    const float* __restrict__ center,
                                    float* __restrict__ out) {
    int d = blockIdx.x * 256 + threadIdx.x;
    float s = 0.f;
#pragma unroll
    for (int y = 0; y < 8; ++y) s += part[y * D_DIM + d];
    out[1 + d] = 0.9f * center[d] + 0.1f * s;
}

// ---- similarity (f32 WMMA) + argmax over teacher patches ----
// One wave per (term, batch, 16-row student tile); 4 waves per block.
// The A tile (reused by all 13 N-tiles) is staged once into LDS in the exact
// per-lane WMMA layout (aT[wave][kstep][lane]); each lane only reads its own
// entries back, so no barrier is needed (same-wave DS ops are in-order).
// Inner loop: per 32-K chunk, 8 batched global B loads + 8 DS A loads, then
// 8 v_wmma_f32_16x16x4_f32 alternating two accumulators (no D->C chain).
__global__ void __launch_bounds__(128, 1)
sim_argmax_kernel(const float* __restrict__ s_fea,
                  const float* __restrict__ t_fea,
                  const float* __restrict__ invn_t,
                  int* __restrict__ idx_out) {
    __shared__ v2f aT[4][64][32];            // 64 KB: 4 waves x 64 ksteps x 32 lanes

    int w    = threadIdx.x >> 5;             // wave in block
    int wave = blockIdx.x * 4 + w;
    int lane = threadIdx.x & 31;
    int half = lane >> 4;
    int lid  = lane & 15;

    int term, b, st;
    if (wave < 832) { term = wave / 416; int r = wave % 416; b = r / 13; st = r % 13; }
    else { int ww = wave - 832; term = 2 + ww / 96; int r = ww % 96; b = r / 3; st = r % 3; }

    int v, iq, sz, soff, poff;
    decode_term(term, v, iq, sz, soff, poff);
    (void)v;

    // A (16x4 f32): lane->M for both halves, half selects the K pair.
    int srow_local = st * 16 + lid;
    int srow = soff + b * sz + ((srow_local < sz) ? srow_local : 0);
    const float* Arow = s_fea + (size_t)srow * F_DIM + 2 * half;

    // stage A fragments into LDS once (each lane's 64 v2f, 16 KB per wave)
#pragma unroll 8
    for (int k = 0; k < 64; ++k)
        aT[w][k][lane] = *(const v2f*)(Arow + k * 4);

    float best[8];
    int   bidx[8];
#pragma unroll
    for (int i = 0; i < 8; ++i) { best[i] = NEG_INF; bidx[i] = 0; }

    for (int nt = 0; nt < 13; ++nt) {        // 13*16 = 208 >= 196 teacher patches
        int p = nt * 16 + lid;
        bool pvalid = (p < P_GL);
        int trow = iq * HALF_T + b * P_GL + (pvalid ? p : 0);
        const float* Brow = t_fea + (size_t)trow * F_DIM + 2 * half;

        v8f acc0 = {}, acc1 = {};
#pragma unroll
        for (int c = 0; c < 8; ++c) {        // 8 chunks x 32 K = 256
            v2f a[8], bb[8];
#pragma unroll
            for (int j = 0; j < 8; ++j)       // batched loads: 8 global + 8 ds
                bb[j] = *(const v2f*)(Brow + c * 32 + j * 4);
#pragma unroll
            for (int j = 0; j < 8; ++j)
                a[j] = aT[w][c * 8 + j][lane];
#pragma unroll
            for (int j = 0; j < 8; j += 2) {
                acc0 = __builtin_amdgcn_wmma_f32_16x16x4_f32(
                    false, a[j],     false, bb[j],     (short)0, acc0, false, false);
                acc1 = __builtin_amdgcn_wmma_f32_16x16x4_f32(
                    false, a[j + 1], false, bb[j + 1], (short)0, acc1, false, false);
            }
        }
        float scale = pvalid ? invn_t[trow] : 0.0f;
#pragma unroll
        for (int i = 0; i < 8; ++i) {
            float val = pvalid ? (acc0[i] + acc1[i]) * scale : NEG_INF;
            if (val > best[i]) { best[i] = val; bidx[i] = p; }
        }
    }
    // reduce across the 16 lanes of each half (first-max tie break: lower p wins)
#pragma unroll
    for (int m = 8; m >= 1; m >>= 1) {
#pragma unroll
        for (int i = 0; i < 8; ++i) {
            float ov = __shfl_xor(best[i], m, 32);
            int   oi = __shfl_xor(bidx[i], m, 32);
            if (ov > best[i] || (ov == best[i] && oi < bidx[i])) {
                best[i] = ov; bidx[i] = oi;
            }
        }
    }
    if (lid == 0) {                    // lane 0 -> rows m=i ; lane 16 -> rows m=i+8
#pragma unroll
        for (int i = 0; i < 8; ++i) {
            int sr = st * 16 + i + 8 * half;
            if (sr < sz) idx_out[poff + b * sz + sr] = bidx[i];
        }
    }
}

// ---- per (term,b,s) pair: -sum_d softmax_t[d] * logp_s[d], weighted ----
__global__ void pair_loss_kernel(const float* __restrict__ s_logits,
                                 const float* __restrict__ t_logits,
                                 const float* __restrict__ center,
                                 const int* __restrict__ epoch,
                                 const float* __restrict__ m_t,
                                 const float* __restrict__ invZ_t,
                                 const float* __restrict__ lse_s,
                                 const int* __restrict__ idx_in,
                                 float* __restrict__ pairloss) {
    __shared__ float sdata[256];
    int pair = blockIdx.x, tid = threadIdx.x;
    int term, pid;
    if (pair < HALF_T)          { term = 0; pid = pair; }
    else if (pair < 2 * HALF_T) { term = 1; pid = pair - HALF_T; }
    else { int w = pair - 2 * HALF_T; term = 2 + w / LOC_ROWS; pid = w % LOC_ROWS; }

    int v, iq, sz, soff, poff;
    decode_term(term, v, iq, sz, soff, poff);
    (void)v;
    int b = pid / sz, s = pid % sz;
    int srow = soff + b * sz + s;
    int p = idx_in[poff + pid];
    p = min(max(p, 0), P_GL - 1);
    int trow = iq * HALF_T + b * P_GL + p;

    float invT = 1.0f / get_ttemp(*epoch);
    float mt = m_t[trow], iz = invZ_t[trow], ls = lse_s[srow];
    const float* srp = s_logits + (size_t)srow * D_DIM;
    const float* trp = t_logits + (size_t)trow * D_DIM;
    float acc = 0.f;
#pragma unroll
    for (int i = 0; i < 16; ++i) {
        int d = tid + i * 256;
        float tv = expf((trp[d] - center[d]) * invT - mt) * iz;
        float sl = srp[d] * 10.0f - ls;
        acc -= tv * sl;
    }
    float tot = blockReduceSum(acc, sdata);
    if (tid == 0) {
        // loss = (1/18) * sum_terms mean_b mean_s
        pairloss[pair] = tot * (1.0f / (18.0f * 32.0f * (float)sz));
    }
}

__global__ void loss_reduce_kernel(const float* __restrict__ pairloss,
                                   float* __restrict__ out) {
    __shared__ float sdata[256];
    float a = 0.f;
    for (int i = threadIdx.x; i < N_PAIRS; i += 256) a += pairloss[i];
    float t = blockReduceSum(a, sdata);
    if (threadIdx.x == 0) out[0] = t;
}

extern "C" void kernel_launch(void* const* d_in, const int* in_sizes, int n_in,
                              void* d_out, int out_size, void* d_ws, size_t ws_size,
                              hipStream_t stream) {
    (void)in_sizes; (void)n_in; (void)out_size; (void)ws_size;
    const float* s_logits = (const float*)d_in[0];
    const float* s_fea    = (const float*)d_in[1];
    const float* t_logits = (const float*)d_in[2];
    const float* t_fea    = (const float*)d_in[3];
    const float* center   = (const float*)d_in[4];
    const int*   epoch    = (const int*)d_in[5];
    float* out = (float*)d_out;

    float* ws     = (float*)d_ws;
    float* invn_t = ws;                          // 12544
    float* m_t    = invn_t + T_ROWS;             // 12544
    float* invZ_t = m_t + T_ROWS;                // 12544
    float* lse_s  = invZ_t + T_ROWS;             // 21760
    float* cpart  = lse_s + S_ROWS;              // 8*4096
    int*   idxbuf = (int*)(cpart + 8 * D_DIM);   // 30976 ints
    float* plos   = (float*)(idxbuf + N_PAIRS);  // 30976

    tfea_invnorm_kernel<<<T_ROWS, 256, 0, stream>>>(t_fea, invn_t);
    tstats_kernel<<<T_ROWS, 256, 0, stream>>>(t_logits, center, epoch, m_t, invZ_t);
    slse_kernel<<<S_ROWS, 256, 0, stream>>>(s_logits, lse_s);
    center_partial_kernel<<<dim3(D_DIM / 256, 8), 256, 0, stream>>>(t_logits, cpart);
    center_final_kernel<<<D_DIM / 256, 256, 0, stream>>>(cpart, center, out);
    sim_argmax_kernel<<<592, 128, 0, stream>>>(s_fea, t_fea, invn_t, idxbuf);
    pair_loss_kernel<<<N_PAIRS, 256, 0, stream>>>(s_logits, t_logits, center, epoch,
                                                  m_t, invZ_t, lse_s, idxbuf, plos);
    loss_reduce_kernel<<<1, 256, 0, stream>>>(plos, out);
}